// DWT2D_6433861009799
// MI455X (gfx1250) — compile-verified
//
#include <hip/hip_runtime.h>

// Haar DWT2 on (16,64,512,512) fp32 -> LL,LH,HL,HH each (16,64,256,256),
// concatenated flat in d_out in that order.
//
// Roofline: 1 GiB in + 1 GiB out, ~0.7 GFLOP -> pure HBM-bound (~92us @ 23.3TB/s).
// Butterfly done via V_WMMA_F32_16X16X4_F32 with the constant 4x4 Haar matrix
// in A rows 0-3; B columns are 16 pixels' (a,b,c,d), which the wave32 B-layout
// lets us load as one coalesced b64 per lane (lanes 0-15 even row, 16-31 odd row).

typedef __attribute__((ext_vector_type(2))) float v2f;
typedef __attribute__((ext_vector_type(8))) float v8f;

__device__ __forceinline__ float haarA(int m, int k) {
  // H rows: 0=LL, 1=LH, 2=HL, 3=HH ; k: 0=a(0,0) 1=b(0,1) 2=c(1,0) 3=d(1,1)
  if (m >= 4) return 0.0f;
  const int dh = (k >> 1) & 1, dw = k & 1;
  const int s = ((m & 1) & dh) ^ (((m >> 1) & 1) & dw);
  return s ? -0.5f : 0.5f;
}

__global__ __launch_bounds__(256) void haar_dwt2_wmma(const float* __restrict__ x,
                                                      float* __restrict__ out,
                                                      int nimg) {
  const int lane = threadIdx.x & 31;
  const int r    = lane & 15;      // pixel column within 16-wide group / A-row
  const int hi   = lane >> 4;      // 0 = even input row (a,b), 1 = odd (c,d)

  const unsigned wave = blockIdx.x * (blockDim.x >> 5) + (threadIdx.x >> 5);
  const unsigned q  = wave & 3;            // quarter of the output row (64 px)
  const unsigned oh = (wave >> 2) & 255;   // output row
  const unsigned n  = wave >> 10;          // image index
  if ((int)n >= nimg) return;              // wave-uniform guard (grid is exact)

  // ---- constant Haar A-matrix (16x4 fp32): VGPR0 = K{0|2}, VGPR1 = K{1|3} ----
  const int k0 = hi ? 2 : 0;
  v2f a;
  a.x = haarA(r, k0);
  a.y = haarA(r, k0 + 1);

  // ---- coalesced B loads: lane<16 -> (a,b) of pixel r, lane>=16 -> (c,d) ----
  const size_t img  = (size_t)n * (512 * 512);
  const float* prow = x + img + (size_t)(2 * oh + hi) * 512 + q * 128 + 2 * r;
  v2f b0 = *(const v2f*)(prow +  0);   // pixels  0..15 of this quarter
  v2f b1 = *(const v2f*)(prow + 32);   // pixels 16..31
  v2f b2 = *(const v2f*)(prow + 64);   // pixels 32..47
  v2f b3 = *(const v2f*)(prow + 96);   // pixels 48..63

  v8f d0, d1, d2, d3;
#if __has_builtin(__builtin_amdgcn_wmma_f32_16x16x4_f32)
  const v8f z = {0.f, 0.f, 0.f, 0.f, 0.f, 0.f, 0.f, 0.f};
  // D rows 0..3 (VGPR elems 0..3 in lanes 0-15) = LL,LH,HL,HH of 16 pixels
  d0 = __builtin_amdgcn_wmma_f32_16x16x4_f32(false, a, false, b0, (short)0, z, false, false);
  d1 = __builtin_amdgcn_wmma_f32_16x16x4_f32(false, a, false, b1, (short)0, z, false, false);
  d2 = __builtin_amdgcn_wmma_f32_16x16x4_f32(false, a, false, b2, (short)0, z, false, false);
  d3 = __builtin_amdgcn_wmma_f32_16x16x4_f32(false, a, false, b3, (short)0, z, false, false);
#else
  // VALU fallback (safety net only): direct scalar butterfly per pixel group.
  const float* pe = x + img + (size_t)(2 * oh) * 512 + q * 128 + 2 * r;
  v8f* dd[4] = {&d0, &d1, &d2, &d3};
  #pragma unroll
  for (int g = 0; g < 4; ++g) {
    const float* p = pe + g * 32;
    float aa = p[0], bb = p[1], cc = p[512], ddv = p[513];
    v8f& D = *dd[g];
    D[0] = (aa + bb + cc + ddv) * 0.5f;
    D[1] = (aa + bb - cc - ddv) * 0.5f;
    D[2] = (aa - bb + cc - ddv) * 0.5f;
    D[3] = (aa - bb - cc + ddv) * 0.5f;
  }
#endif

  // ---- stores: lanes 0-15 hold valid rows M=0..3; contiguous per plane ----
  if (!hi) {
    const size_t plane = (size_t)nimg * (256 * 256);
    const size_t ob = (size_t)n * (256 * 256) + (size_t)oh * 256 + q * 64 + r;
    float* oLL = out + ob;
    float* oLH = out + plane + ob;
    float* oHL = out + 2 * plane + ob;
    float* oHH = out + 3 * plane + ob;
    oLL[0] = d0[0]; oLL[16] = d1[0]; oLL[32] = d2[0]; oLL[48] = d3[0];
    oLH[0] = d0[1]; oLH[16] = d1[1]; oLH[32] = d2[1]; oLH[48] = d3[1];
    oHL[0] = d0[2]; oHL[16] = d1[2]; oHL[32] = d2[2]; oHL[48] = d3[2];
    oHH[0] = d0[3]; oHH[16] = d1[3]; oHH[32] = d2[3]; oHH[48] = d3[3];
  }
}

extern "C" void kernel_launch(void* const* d_in, const int* in_sizes, int n_in,
                              void* d_out, int out_size, void* d_ws, size_t ws_size,
                              hipStream_t stream) {
  const float* x = (const float*)d_in[0];
  float* out = (float*)d_out;
  const int nimg = in_sizes[0] / (512 * 512);      // 16*64 = 1024 images of 512x512
  // waves = nimg * 256 rows * 4 quarters ; 8 waves per 256-thread block
  const int blocks = nimg * 128;
  haar_dwt2_wmma<<<blocks, 256, 0, stream>>>(x, out, nimg);
}